// ResidualTemporalRelationalGraphConv_17291538334035
// MI455X (gfx1250) — compile-verified
//
#include <hip/hip_runtime.h>
#include <math.h>

// Problem constants (match reference)
constexpr int CN_SRC = 50000;
constexpr int CN_DST = 10000;
constexpr int CE     = 250000;
constexpr int CR     = 4;
constexpr int CIN    = 256;
constexpr int COUT   = 256;
constexpr int CH     = 4;
constexpr int CDH    = 64;
constexpr float NEG_SLOPE = 0.2f;

typedef __attribute__((ext_vector_type(2))) float v2f;
typedef __attribute__((ext_vector_type(8))) float v8f;

// ---------------------------------------------------------------------------
// WMMA f32 GEMM:  C[M x 256] = A[M x 256] * B[256 x 256], row-major.
// Each wave computes a 16(M) x 64(N) tile = 4 accumulators; the A-fragment is
// loaded once per K-step and reused by 4 V_WMMA_F32_16X16X4_F32 ops (4x better
// A reuse than 1 tile/wave). A/B/C layouts per CDNA5 ISA 7.12.2:
//   A-frag (16x4 f32): lanes 0-15 hold M=0..15 K={k0,k0+1}; lanes 16-31 K={k0+2,k0+3}
//   C/D (16x16 f32): VGPR i -> M=i (lanes 0-15), M=i+8 (lanes 16-31), N=lane%16
// ---------------------------------------------------------------------------
__global__ void rgat_gemm_wmma_f32(const float* __restrict__ A,
                                   const float* __restrict__ B,
                                   float* __restrict__ C, int M) {
    const int N = COUT, K = CIN;
    const int lane = threadIdx.x & 31;
    const int wave = threadIdx.x >> 5;
    const int groupsN = N / 64;                       // 4 N-groups of 64
    const int nGroups = (M / 16) * groupsN;
    const int group = blockIdx.x * (blockDim.x >> 5) + wave;
    if (group >= nGroups) return;                     // wave-uniform: EXEC all-1s

    const int tm = group / groupsN;
    const int tg = group % groupsN;
    const int m     = tm * 16 + (lane & 15);
    const int nbase = tg * 64 + (lane & 15);
    const int koff  = (lane >> 4) << 1;               // 0 or 2

    v8f acc0 = {}, acc1 = {}, acc2 = {}, acc3 = {};
    #pragma unroll 4
    for (int k0 = 0; k0 < K; k0 += 4) {
        v2f a;
        const float* ap = A + (size_t)m * K + (k0 + koff);
        a.x = ap[0];
        a.y = ap[1];
        const float* bp = B + (size_t)(k0 + koff) * N + nbase;
        v2f b0, b1, b2, b3;
        b0.x = bp[0];      b0.y = bp[N];
        b1.x = bp[16];     b1.y = bp[N + 16];
        b2.x = bp[32];     b2.y = bp[N + 32];
        b3.x = bp[48];     b3.y = bp[N + 48];
        acc0 = __builtin_amdgcn_wmma_f32_16x16x4_f32(false, a, false, b0, (short)0, acc0, false, false);
        acc1 = __builtin_amdgcn_wmma_f32_16x16x4_f32(false, a, false, b1, (short)0, acc1, false, false);
        acc2 = __builtin_amdgcn_wmma_f32_16x16x4_f32(false, a, false, b2, (short)0, acc2, false, false);
        acc3 = __builtin_amdgcn_wmma_f32_16x16x4_f32(false, a, false, b3, (short)0, acc3, false, false);
    }

    const int rbase = tm * 16 + ((lane >> 4) << 3);
    #pragma unroll
    for (int i = 0; i < 8; ++i) {
        float* cp = C + (size_t)(rbase + i) * N + nbase;
        cp[0]  = acc0[i];
        cp[16] = acc1[i];
        cp[32] = acc2[i];
        cp[48] = acc3[i];
    }
}

// ---------------------------------------------------------------------------
// el[node,h] = sum_d h[node,h,d]*al[h,d];  er (only node < N_DST) with ar.
// One wave per (node, head); wave32 shuffle reduction.
// ---------------------------------------------------------------------------
__global__ void rgat_elr(const float* __restrict__ h,
                         const float* __restrict__ al,
                         const float* __restrict__ ar,
                         float* __restrict__ el, float* __restrict__ er) {
    const int warp = (blockIdx.x * blockDim.x + threadIdx.x) >> 5;
    const int lane = threadIdx.x & 31;
    if (warp >= CN_SRC * CH) return;
    const int node = warp >> 2;
    const int head = warp & 3;
    const float* hp = h + (size_t)node * COUT + head * CDH;
    const float h0 = hp[lane], h1 = hp[lane + 32];
    float sl = h0 * al[head * CDH + lane] + h1 * al[head * CDH + lane + 32];
    const bool doR = node < CN_DST;
    float sr = 0.f;
    if (doR)
        sr = h0 * ar[head * CDH + lane] + h1 * ar[head * CDH + lane + 32];
    #pragma unroll
    for (int off = 16; off > 0; off >>= 1) {
        sl += __shfl_xor(sl, off, 32);
        sr += __shfl_xor(sr, off, 32);
    }
    if (lane == 0) {
        el[warp] = sl;
        if (doR) er[node * CH + head] = sr;
    }
}

__global__ void rgat_fill(float* __restrict__ p, float v, int n) {
    int i = blockIdx.x * blockDim.x + threadIdx.x;
    if (i < n) p[i] = v;
}

__device__ inline void atomicMaxFloat(float* addr, float v) {
    if (v >= 0.0f) atomicMax((int*)addr, __float_as_int(v));
    else           atomicMin((unsigned int*)addr, __float_as_uint(v));
}

// e = leaky_relu(el[src]+er[dst]); store e; emax[dst] = max(emax, e) per head.
__global__ void rgat_edge_score(const int* __restrict__ src,
                                const int* __restrict__ dst,
                                const float* __restrict__ el,
                                const float* __restrict__ er,
                                float* __restrict__ ebuf,
                                float* __restrict__ emax) {
    int i = blockIdx.x * blockDim.x + threadIdx.x;
    if (i >= CE) return;
    const int s = src[i], d = dst[i];
    float4 l  = ((const float4*)el)[s];
    float4 rr = ((const float4*)er)[d];
    float4 e;
    e.x = l.x + rr.x; e.x = e.x > 0.f ? e.x : NEG_SLOPE * e.x;
    e.y = l.y + rr.y; e.y = e.y > 0.f ? e.y : NEG_SLOPE * e.y;
    e.z = l.z + rr.z; e.z = e.z > 0.f ? e.z : NEG_SLOPE * e.z;
    e.w = l.w + rr.w; e.w = e.w > 0.f ? e.w : NEG_SLOPE * e.w;
    ((float4*)ebuf)[i] = e;
    float* em = emax + (size_t)d * CH;
    atomicMaxFloat(em + 0, e.x);
    atomicMaxFloat(em + 1, e.y);
    atomicMaxFloat(em + 2, e.z);
    atomicMaxFloat(em + 3, e.w);
}

// emax = finite ? emax : 0   (empty segments)
__global__ void rgat_fix_emax(float* __restrict__ emax, int n) {
    int i = blockIdx.x * blockDim.x + threadIdx.x;
    if (i >= n) return;
    float v = emax[i];
    if (!(v >= -3.0e38f && v <= 3.0e38f)) emax[i] = 0.0f;
}

// Warp per edge: ex = exp(e - emax[dst]); denom[dst] += ex;
// aggnum[dst, :] += ex[head] * h[src, :]   (softmax numerator, fused)
__global__ void rgat_edge_accum(const int* __restrict__ src,
                                const int* __restrict__ dst,
                                const float* __restrict__ ebuf,
                                const float* __restrict__ emax,
                                const float* __restrict__ h,
                                float* __restrict__ denom,
                                float* __restrict__ aggnum) {
    const int warp = (blockIdx.x * blockDim.x + threadIdx.x) >> 5;
    const int lane = threadIdx.x & 31;
    if (warp >= CE) return;
    const int s = src[warp], d = dst[warp];
    float4 e  = ((const float4*)ebuf)[warp];
    float4 mx = ((const float4*)emax)[d];
    const float ex0 = expf(e.x - mx.x);
    const float ex1 = expf(e.y - mx.y);
    const float ex2 = expf(e.z - mx.z);
    const float ex3 = expf(e.w - mx.w);
    if (lane < 4) {
        float exl = (lane == 0) ? ex0 : (lane == 1) ? ex1 : (lane == 2) ? ex2 : ex3;
        atomicAdd(&denom[(size_t)d * CH + lane], exl);
    }
    const float* hp = h + (size_t)s * COUT;
    float* ap = aggnum + (size_t)d * COUT;
    #pragma unroll
    for (int it = 0; it < 8; ++it) {
        const int c = lane + (it << 5);
        const float exh = (it < 2) ? ex0 : (it < 4) ? ex1 : (it < 6) ? ex2 : ex3;
        atomicAdd(&ap[c], exh * hp[c]);
    }
}

// agg += aggnum / max(denom, 1e-9)  (per destination, per head)
__global__ void rgat_rel_final(float* __restrict__ agg,
                               const float* __restrict__ aggnum,
                               const float* __restrict__ denom) {
    int t = blockIdx.x * blockDim.x + threadIdx.x;
    if (t >= CN_DST * COUT) return;
    const int node = t >> 8;
    const int head = (t & 255) >> 6;
    agg[t] += aggnum[t] / fmaxf(denom[node * CH + head], 1e-9f);
}

// out = relu(hloop + loop_b + agg / R)
__global__ void rgat_epilogue(float* __restrict__ out,
                              const float* __restrict__ hloop,
                              const float* __restrict__ loop_b,
                              const float* __restrict__ agg) {
    int t = blockIdx.x * blockDim.x + threadIdx.x;
    if (t >= CN_DST * COUT) return;
    const int c = t & 255;
    float v = hloop[t] + loop_b[c] + agg[t] * (1.0f / CR);
    out[t] = v > 0.f ? v : 0.f;
}

// ---------------------------------------------------------------------------
extern "C" void kernel_launch(void* const* d_in, const int* in_sizes, int n_in,
                              void* d_out, int out_size, void* d_ws, size_t ws_size,
                              hipStream_t stream) {
    const float* x       = (const float*)d_in[0];
    const float* W       = (const float*)d_in[1];
    const float* attn_l  = (const float*)d_in[2];
    const float* attn_r  = (const float*)d_in[3];
    const float* loop_w  = (const float*)d_in[4];
    const float* loop_b  = (const float*)d_in[5];
    const int*   edge_src = (const int*)d_in[6];
    const int*   edge_dst = (const int*)d_in[7];

    float* ws    = (float*)d_ws;
    float* h     = ws;                              // 50000*256
    float* el    = h     + (size_t)CN_SRC * COUT;   // 50000*4
    float* er    = el    + (size_t)CN_SRC * CH;     // 10000*4
    float* ebuf  = er    + (size_t)CN_DST * CH;     // 250000*4
    float* emax  = ebuf  + (size_t)CE * CH;         // 10000*4
    float* denom = emax  + (size_t)CN_DST * CH;     // 10000*4
    float* aggnum= denom + (size_t)CN_DST * CH;     // 10000*256
    float* agg   = aggnum+ (size_t)CN_DST * COUT;   // 10000*256
    float* hloop = agg   + (size_t)CN_DST * COUT;   // 10000*256

    const int TB = 256;
    const int ndElems = CN_DST * COUT;

    rgat_fill<<<(ndElems + TB - 1) / TB, TB, 0, stream>>>(agg, 0.0f, ndElems);

    for (int r = 0; r < CR; ++r) {
        // h = x @ W[r]   (WMMA, 16x64 tile per wave, 8 waves/block)
        {
            int nGroups = (CN_SRC / 16) * (COUT / 64);
            int blocks = (nGroups + 7) / 8;
            rgat_gemm_wmma_f32<<<blocks, TB, 0, stream>>>(
                x, W + (size_t)r * CIN * COUT, h, CN_SRC);
        }
        // attention scores el / er
        {
            int warps = CN_SRC * CH;
            rgat_elr<<<(warps + 7) / 8, TB, 0, stream>>>(
                h, attn_l + (size_t)r * CH * CDH, attn_r + (size_t)r * CH * CDH, el, er);
        }
        // init per-relation accumulators
        rgat_fill<<<(CN_DST * CH + TB - 1) / TB, TB, 0, stream>>>(emax, -INFINITY, CN_DST * CH);
        rgat_fill<<<(CN_DST * CH + TB - 1) / TB, TB, 0, stream>>>(denom, 0.0f, CN_DST * CH);
        rgat_fill<<<(ndElems + TB - 1) / TB, TB, 0, stream>>>(aggnum, 0.0f, ndElems);
        // edge pass 1: scores + segment max
        rgat_edge_score<<<(CE + TB - 1) / TB, TB, 0, stream>>>(
            edge_src + (size_t)r * CE, edge_dst + (size_t)r * CE, el, er, ebuf, emax);
        rgat_fix_emax<<<(CN_DST * CH + TB - 1) / TB, TB, 0, stream>>>(emax, CN_DST * CH);
        // edge pass 2: exp + denom + fused numerator scatter (warp per edge)
        rgat_edge_accum<<<(CE + 7) / 8, TB, 0, stream>>>(
            edge_src + (size_t)r * CE, edge_dst + (size_t)r * CE,
            ebuf, emax, h, denom, aggnum);
        // agg += aggnum / denom
        rgat_rel_final<<<(ndElems + TB - 1) / TB, TB, 0, stream>>>(agg, aggnum, denom);
    }

    // self-loop GEMM: hloop = x[:N_DST] @ loop_w
    {
        int nGroups = (CN_DST / 16) * (COUT / 64);
        int blocks = (nGroups + 7) / 8;
        rgat_gemm_wmma_f32<<<blocks, TB, 0, stream>>>(x, loop_w, hloop, CN_DST);
    }
    // out = relu(hloop + b + agg/4)
    rgat_epilogue<<<(ndElems + TB - 1) / TB, TB, 0, stream>>>(
        (float*)d_out, hloop, loop_b, agg);
}